// TopKPoolBroadcastGCN_51247549776544
// MI455X (gfx1250) — compile-verified
//
#include <hip/hip_runtime.h>
#include <hip/hip_bf16.h>

#define IND   512
#define HID   256
#define OUTD  256
#define KPOOL 4096

typedef __attribute__((ext_vector_type(2))) float v2f;
typedef __attribute__((ext_vector_type(8))) float v8f;
typedef unsigned long long ull;

__device__ __forceinline__ unsigned mono_f32(float f) {
  unsigned u = __float_as_uint(f);
  return (u & 0x80000000u) ? ~u : (u | 0x80000000u);
}

// ------------------------------------------------------------------
// WMMA f32 GEMM: C[M,N] = A[M,K] * B[K,N].  One wave -> 16x64 tile.
// Layouts per CDNA5 ISA 7.12.2:
//   A 16x4 f32 : lane holds row M=lane&15, K pair = (lane>>4)*2
//   B 4x16 f32 : lane holds col N=lane&15, K rows (lane>>4)*2, +1
//   C 16x16 f32: vgpr r, lanes0-15 -> M=r, lanes16-31 -> M=r+8
// ------------------------------------------------------------------
template <typename TA>
__global__ void __launch_bounds__(128)
wmma_gemm(const TA* __restrict__ A, const float* __restrict__ B,
          float* __restrict__ C, int Mtiles, int K,
          int lda, int ldb, int ldc) {
  const int wave = threadIdx.x >> 5;
  const int lane = threadIdx.x & 31;
  const int mtile = blockIdx.x * 4 + wave;
  if (mtile >= Mtiles) return;
  const int m0 = mtile * 16;
  const int n0 = blockIdx.y * 64;
  const int nc = lane & 15;           // A row within tile == B/C column
  const int kh = (lane >> 4) * 2;     // K sub-offset 0 or 2

  v8f acc0 = {}, acc1 = {}, acc2 = {}, acc3 = {};
  const TA* arow = A + (size_t)(m0 + nc) * lda + kh;
  const float* bcol = B + n0 + nc;

  for (int k = 0; k < K; k += 4) {
    v2f a;
    a.x = (float)arow[k];
    a.y = (float)arow[k + 1];
    const float* bp = bcol + (size_t)(k + kh) * ldb;
    v2f b0, b1, b2, b3;
    b0.x = bp[0];  b0.y = bp[ldb];
    b1.x = bp[16]; b1.y = bp[ldb + 16];
    b2.x = bp[32]; b2.y = bp[ldb + 32];
    b3.x = bp[48]; b3.y = bp[ldb + 48];
    acc0 = __builtin_amdgcn_wmma_f32_16x16x4_f32(false, a, false, b0, (short)0, acc0, false, false);
    acc1 = __builtin_amdgcn_wmma_f32_16x16x4_f32(false, a, false, b1, (short)0, acc1, false, false);
    acc2 = __builtin_amdgcn_wmma_f32_16x16x4_f32(false, a, false, b2, (short)0, acc2, false, false);
    acc3 = __builtin_amdgcn_wmma_f32_16x16x4_f32(false, a, false, b3, (short)0, acc3, false, false);
  }

  const int rbase = m0 + ((lane >> 4) ? 8 : 0);
  float* crow = C + (size_t)rbase * ldc + n0 + nc;
#pragma unroll
  for (int r = 0; r < 8; ++r) {
    crow[(size_t)r * ldc + 0]  = acc0[r];
    crow[(size_t)r * ldc + 16] = acc1[r];
    crow[(size_t)r * ldc + 32] = acc2[r];
    crow[(size_t)r * ldc + 48] = acc3[r];
  }
}

// ------------------------- utility kernels -------------------------
__global__ void zero_u32(unsigned* p, size_t n) {
  size_t i = (size_t)blockIdx.x * blockDim.x + threadIdx.x;
  size_t stride = (size_t)gridDim.x * blockDim.x;
  for (; i < n; i += stride) p[i] = 0u;
}

__global__ void small_init(unsigned* state, ull* bgenc, int* bgidx, int kneed) {
  state[0] = 0u;                 // prefix
  state[1] = (unsigned)kneed;    // remaining to select
  state[2] = 0u;                 // tie counter
  *bgenc = 0ull;
  *bgidx = 0x7fffffff;
}

// ----------------------- degrees & GCN agg -------------------------
__global__ void deg_init(float* degA, int* degS, int n) {
  int i = blockIdx.x * blockDim.x + threadIdx.x;
  if (i < n) { degA[i] = 1.0f; degS[i] = 0; }   // self-loop for GCN degree
}

__global__ void deg_accum(const int* __restrict__ src, const int* __restrict__ dst,
                          float* degA, int* degS, int E) {
  int e = blockIdx.x * blockDim.x + threadIdx.x;
  if (e < E) {
    atomicAdd(&degA[dst[e]], 1.0f);
    atomicAdd(&degS[src[e]], 1);
  }
}

__global__ void compute_dinv(const float* degA, float* dinv, int n) {
  int i = blockIdx.x * blockDim.x + threadIdx.x;
  if (i < n) dinv[i] = rsqrtf(degA[i]);
}

__global__ void gcn_selfloop(const float* __restrict__ h, const float* __restrict__ dinv,
                             float* __restrict__ x1, int n) {
  int i = blockIdx.x;
  if (i >= n) return;
  float c = dinv[i] * dinv[i];
  x1[(size_t)i * HID + threadIdx.x] = c * h[(size_t)i * HID + threadIdx.x];
}

__global__ void gcn_edge_agg(const float* __restrict__ h, const int* __restrict__ src,
                             const int* __restrict__ dst, const float* __restrict__ dinv,
                             float* __restrict__ x1, int E) {
  int e = blockIdx.x;
  if (e >= E) return;
  int s = src[e], d = dst[e];
  float c = dinv[s] * dinv[d];
  atomicAdd(&x1[(size_t)d * HID + threadIdx.x], c * h[(size_t)s * HID + threadIdx.x]);
}

__global__ void gcn_finalize(float* __restrict__ x1, const float* __restrict__ b1,
                             const float* __restrict__ wsc, float* __restrict__ raw, int n) {
  __shared__ float red[HID];
  int i = blockIdx.x;
  if (i >= n) return;
  float v = x1[(size_t)i * HID + threadIdx.x] + b1[threadIdx.x];
  v = fmaxf(v, 0.0f);
  x1[(size_t)i * HID + threadIdx.x] = v;
  red[threadIdx.x] = v * wsc[threadIdx.x];
  __syncthreads();
  for (int o = HID / 2; o > 0; o >>= 1) {
    if (threadIdx.x < o) red[threadIdx.x] += red[threadIdx.x + o];
    __syncthreads();
  }
  if (threadIdx.x == 0) raw[i] = red[0];
}

// -------------------- radix top-K select (K-th largest) ------------
__global__ void rs_hist(const float* __restrict__ raw, int n,
                        const unsigned* __restrict__ state, unsigned* hist, int shift) {
  __shared__ unsigned lh[256];
  lh[threadIdx.x] = 0u;
  __syncthreads();
  unsigned prefix = state[0];
  unsigned maskHi = (shift + 8 >= 32) ? 0u : (0xFFFFFFFFu << (shift + 8));
  for (int i = blockIdx.x * blockDim.x + threadIdx.x; i < n; i += gridDim.x * blockDim.x) {
    unsigned u = mono_f32(raw[i]);
    if ((u & maskHi) == (prefix & maskHi))
      atomicAdd(&lh[(u >> shift) & 255u], 1u);
  }
  __syncthreads();
  atomicAdd(&hist[threadIdx.x], lh[threadIdx.x]);
}

__global__ void rs_pick(unsigned* state, unsigned* hist, int shift) {
  __shared__ unsigned h[256];
  h[threadIdx.x] = hist[threadIdx.x];
  __syncthreads();
  if (threadIdx.x == 0) {
    unsigned need = state[1];
    unsigned cum = 0u;
    int b = 255;
    while (b > 0) {
      unsigned c = h[b];
      if (cum + c >= need) break;
      cum += c;
      --b;
    }
    state[0] |= ((unsigned)b) << shift;
    state[1] = need - cum;
  }
  hist[threadIdx.x] = 0u;   // ready for next pass
}

__global__ void keep_mark(const float* __restrict__ raw, unsigned* state,
                          int* __restrict__ keep, int n) {
  int i = blockIdx.x * blockDim.x + threadIdx.x;
  if (i >= n) return;
  unsigned pivot = state[0];
  unsigned need = state[1];
  unsigned u = mono_f32(raw[i]);
  int k = 0;
  if (u > pivot) k = 1;
  else if (u == pivot) {
    unsigned t = atomicAdd(&state[2], 1u);
    if (t < need) k = 1;
  }
  keep[i] = k;
}

// rank (deterministic prefix sum over keep) -> cid for kept, -1 else
__global__ void __launch_bounds__(1024)
rank_scan(const int* __restrict__ keep, int* __restrict__ cid, int n) {
  __shared__ int sdata[1024];
  __shared__ int base;
  if (threadIdx.x == 0) base = 0;
  __syncthreads();
  for (int start = 0; start < n; start += 1024) {
    int i = start + (int)threadIdx.x;
    int v = (i < n) ? keep[i] : 0;
    sdata[threadIdx.x] = v;
    __syncthreads();
    for (int off = 1; off < 1024; off <<= 1) {
      int t = (threadIdx.x >= (unsigned)off) ? sdata[threadIdx.x - off] : 0;
      __syncthreads();
      sdata[threadIdx.x] += t;
      __syncthreads();
    }
    int incl = sdata[threadIdx.x];
    if (i < n) cid[i] = v ? (base + incl - 1) : -1;
    __syncthreads();
    if (threadIdx.x == 1023) base += incl;
    __syncthreads();
  }
}

// ----------------- cluster assignment for dropped nodes ------------
__global__ void best_edges(const int* __restrict__ src, const int* __restrict__ dst,
                           const int* __restrict__ keep, const int* __restrict__ degS,
                           ull* __restrict__ best, int E, int n) {
  int e = blockIdx.x * blockDim.x + threadIdx.x;
  if (e >= E) return;
  int s = src[e], d = dst[e];
  if (keep[d]) {
    ull cand = (((ull)(unsigned)(degS[d] + 1)) << 32) | (unsigned)(n - 1 - d);
    atomicMax(&best[s], cand);
  }
  if (keep[s]) {
    ull cand = (((ull)(unsigned)(degS[s] + 1)) << 32) | (unsigned)(n - 1 - s);
    atomicMax(&best[d], cand);
  }
}

__global__ void bg_pass_a(const int* keep, const int* degS, const float* raw,
                          ull* bgenc, int n) {
  int i = blockIdx.x * blockDim.x + threadIdx.x;
  if (i >= n || !keep[i]) return;
  ull enc = (((ull)(unsigned)degS[i]) << 32) | mono_f32(raw[i]);
  atomicMax(bgenc, enc);
}

__global__ void bg_pass_b(const int* keep, const int* degS, const float* raw,
                          const ull* bgenc, int* bgidx, int n) {
  int i = blockIdx.x * blockDim.x + threadIdx.x;
  if (i >= n || !keep[i]) return;
  ull enc = (((ull)(unsigned)degS[i]) << 32) | mono_f32(raw[i]);
  if (enc == *bgenc) atomicMin(bgidx, i);
}

__global__ void cid_finalize(const int* __restrict__ keep, const ull* __restrict__ best,
                             const int* __restrict__ bgidx, int* __restrict__ cid, int n) {
  int i = blockIdx.x * blockDim.x + threadIdx.x;
  if (i >= n) return;
  if (!keep[i]) {                    // writes only to non-kept; reads only kept entries
    ull b = best[i];
    int c;
    if (b != 0ull) {
      int t = (n - 1) - (int)(unsigned)(b & 0xffffffffu);
      c = cid[t];
    } else {
      c = cid[*bgidx];
    }
    cid[i] = c;
  }
}

// --------------------------- pooling -------------------------------
__global__ void pool_accum(const float* __restrict__ x1, const float* __restrict__ raw,
                           const int* __restrict__ cid, float* __restrict__ sums,
                           float* __restrict__ cnt, int n) {
  int i = blockIdx.x;
  if (i >= n) return;
  int c = cid[i];
  float g = tanhf(raw[i]);
  atomicAdd(&sums[(size_t)c * HID + threadIdx.x], x1[(size_t)i * HID + threadIdx.x] * g);
  if (threadIdx.x == 0) atomicAdd(&cnt[c], 1.0f);
}

__global__ void pool_div(const float* sums, const float* cnt, float* xp) {
  int j = blockIdx.x;
  xp[(size_t)j * HID + threadIdx.x] =
      sums[(size_t)j * HID + threadIdx.x] / fmaxf(cnt[j], 1.0f);
}

// ----------------------- pooled adjacency --------------------------
__global__ void build_A(const int* __restrict__ src, const int* __restrict__ dst,
                        const int* __restrict__ cid, unsigned char* __restrict__ Ab, int E) {
  int e = blockIdx.x * blockDim.x + threadIdx.x;
  if (e >= E) return;
  int cu = cid[src[e]], cv = cid[dst[e]];
  if (cu != cv) Ab[(size_t)cv * KPOOL + cu] = (unsigned char)1;
}

__global__ void rowdeg(const unsigned char* __restrict__ Ab, float* __restrict__ di) {
  __shared__ int red[256];
  int r = blockIdx.x;
  int s = 0;
  for (int c = threadIdx.x; c < KPOOL; c += 256) s += Ab[(size_t)r * KPOOL + c];
  red[threadIdx.x] = s;
  __syncthreads();
  for (int o = 128; o > 0; o >>= 1) {
    if (threadIdx.x < o) red[threadIdx.x] += red[threadIdx.x + o];
    __syncthreads();
  }
  if (threadIdx.x == 0) di[r] = rsqrtf(1.0f + (float)red[0]);   // +1 self loop
}

__global__ void scale_rows(const float* __restrict__ Y, const float* __restrict__ di,
                           float* __restrict__ Y2) {
  int j = blockIdx.x;
  Y2[(size_t)j * HID + threadIdx.x] = di[j] * Y[(size_t)j * HID + threadIdx.x];
}

// xp2 = di[r]*Z + di[r]^2*Y + b2
__global__ void xp2_combine(const float* __restrict__ Z, const float* __restrict__ Y,
                            const float* __restrict__ di, const float* __restrict__ b2,
                            float* __restrict__ xp2) {
  int r = blockIdx.x;
  float d = di[r];
  xp2[(size_t)r * HID + threadIdx.x] =
      d * Z[(size_t)r * HID + threadIdx.x] +
      d * d * Y[(size_t)r * HID + threadIdx.x] + b2[threadIdx.x];
}

// out += xp2[cid] + b_skip   (out already holds x1 @ W_skip)
__global__ void final_add(float* __restrict__ out, const float* __restrict__ xp2,
                          const int* __restrict__ cid, const float* __restrict__ bsk, int n) {
  int i = blockIdx.x;
  if (i >= n) return;
  int c = cid[i];
  out[(size_t)i * OUTD + threadIdx.x] +=
      xp2[(size_t)c * OUTD + threadIdx.x] + bsk[threadIdx.x];
}

// ===================================================================
extern "C" void kernel_launch(void* const* d_in, const int* in_sizes, int n_in,
                              void* d_out, int out_size, void* d_ws, size_t ws_size,
                              hipStream_t stream) {
  (void)n_in; (void)out_size; (void)ws_size;
  const float* x   = (const float*)d_in[0];
  const int*   ei  = (const int*)d_in[1];
  const float* W1  = (const float*)d_in[2];
  const float* b1  = (const float*)d_in[3];
  const float* W2  = (const float*)d_in[4];
  const float* b2  = (const float*)d_in[5];
  const float* wsc = (const float*)d_in[6];
  const float* Wsk = (const float*)d_in[7];
  const float* bsk = (const float*)d_in[8];

  const int N = in_sizes[0] / IND;
  const int E = in_sizes[1] / 2;
  const int* src = ei;
  const int* dst = ei + E;

  // ------- workspace layout -------
  char* wsb = (char*)d_ws;
  size_t off = 0;
  auto alloc = [&](size_t bytes) -> void* {
    void* p = wsb + off;
    off += (bytes + 255) & ~(size_t)255;
    return p;
  };
  float* h     = (float*)alloc((size_t)N * HID * 4);
  float* x1    = (float*)alloc((size_t)N * HID * 4);
  float* raw   = (float*)alloc((size_t)N * 4);
  float* degA  = (float*)alloc((size_t)N * 4);
  int*   degS  = (int*)alloc((size_t)N * 4);
  float* dinv  = (float*)alloc((size_t)N * 4);
  int*   keep  = (int*)alloc((size_t)N * 4);
  int*   cid   = (int*)alloc((size_t)N * 4);
  ull*   best  = (ull*)alloc((size_t)N * 8);
  unsigned* state = (unsigned*)alloc(64);
  unsigned* hist  = (unsigned*)alloc(256 * 4);
  ull*   bgenc = (ull*)alloc(16);
  int*   bgidx = (int*)((char*)bgenc + 8);
  float* sums  = (float*)alloc((size_t)KPOOL * HID * 4);
  float* cnt   = (float*)alloc((size_t)KPOOL * 4);
  float* xp    = (float*)alloc((size_t)KPOOL * HID * 4);
  unsigned char* Ab = (unsigned char*)alloc((size_t)KPOOL * KPOOL);
  float* di    = (float*)alloc((size_t)KPOOL * 4);
  float* Y     = (float*)alloc((size_t)KPOOL * HID * 4);
  float* Y2    = (float*)alloc((size_t)KPOOL * HID * 4);
  float* Z     = (float*)alloc((size_t)KPOOL * HID * 4);
  float* xp2   = (float*)alloc((size_t)KPOOL * HID * 4);
  float* out   = (float*)d_out;

  const int Mt_nodes = N / 16;        // 50000/16 = 3125 exactly
  const int Mt_pool  = KPOOL / 16;    // 256
  const int TB = 256;
  const int gN = (N + TB - 1) / TB;
  const int gE = (E + TB - 1) / TB;

  // 1) h = x @ W1  (K=512)
  wmma_gemm<float><<<dim3((Mt_nodes + 3) / 4, HID / 64), 128, 0, stream>>>(
      x, W1, h, Mt_nodes, IND, IND, HID, HID);

  // 2) degrees (GCN dst-degree w/ self loop; out-degree for clustering)
  deg_init<<<gN, TB, 0, stream>>>(degA, degS, N);
  deg_accum<<<gE, TB, 0, stream>>>(src, dst, degA, degS, E);
  compute_dinv<<<gN, TB, 0, stream>>>(degA, dinv, N);

  // 3) GCN aggregation: self loop + edges, then relu/bias/score
  gcn_selfloop<<<N, HID, 0, stream>>>(h, dinv, x1, N);
  gcn_edge_agg<<<E, HID, 0, stream>>>(h, src, dst, dinv, x1, E);
  gcn_finalize<<<N, HID, 0, stream>>>(x1, b1, wsc, raw, N);

  // 4) top-K radix select on raw
  small_init<<<1, 1, 0, stream>>>(state, bgenc, bgidx, KPOOL);
  zero_u32<<<1, 256, 0, stream>>>(hist, 256);
  const int shifts[4] = {24, 16, 8, 0};
  for (int p = 0; p < 4; ++p) {
    rs_hist<<<256, 256, 0, stream>>>(raw, N, state, hist, shifts[p]);
    rs_pick<<<1, 256, 0, stream>>>(state, hist, shifts[p]);
  }
  keep_mark<<<gN, TB, 0, stream>>>(raw, state, keep, N);
  rank_scan<<<1, 1024, 0, stream>>>(keep, cid, N);

  // 5) assign dropped nodes to best kept neighbor (fallback: global best)
  zero_u32<<<256, 256, 0, stream>>>((unsigned*)best, (size_t)N * 2);
  best_edges<<<gE, TB, 0, stream>>>(src, dst, keep, degS, best, E, N);
  bg_pass_a<<<gN, TB, 0, stream>>>(keep, degS, raw, bgenc, N);
  bg_pass_b<<<gN, TB, 0, stream>>>(keep, degS, raw, bgenc, bgidx, N);
  cid_finalize<<<gN, TB, 0, stream>>>(keep, best, bgidx, cid, N);

  // 6) mean pooling of gated features
  zero_u32<<<512, 256, 0, stream>>>((unsigned*)sums, (size_t)KPOOL * HID);
  zero_u32<<<16, 256, 0, stream>>>((unsigned*)cnt, (size_t)KPOOL);
  pool_accum<<<N, HID, 0, stream>>>(x1, raw, cid, sums, cnt, N);
  pool_div<<<KPOOL, HID, 0, stream>>>(sums, cnt, xp);

  // 7) pooled adjacency + normalization
  zero_u32<<<1024, 256, 0, stream>>>((unsigned*)Ab, (size_t)KPOOL * KPOOL / 4);
  build_A<<<gE, TB, 0, stream>>>(src, dst, cid, Ab, E);
  rowdeg<<<KPOOL, 256, 0, stream>>>(Ab, di);

  // 8) conv2 on pooled graph: Y = xp@W2; Z = A@(di*Y); xp2 = di*Z + di^2*Y + b2
  wmma_gemm<float><<<dim3(Mt_pool / 4, HID / 64), 128, 0, stream>>>(
      xp, W2, Y, Mt_pool, HID, HID, HID, HID);
  scale_rows<<<KPOOL, HID, 0, stream>>>(Y, di, Y2);
  wmma_gemm<unsigned char><<<dim3(Mt_pool / 4, HID / 64), 128, 0, stream>>>(
      Ab, Y2, Z, Mt_pool, KPOOL, KPOOL, HID, HID);
  xp2_combine<<<KPOOL, HID, 0, stream>>>(Z, Y, di, b2, xp2);

  // 9) skip connection + broadcast back
  wmma_gemm<float><<<dim3((Mt_nodes + 3) / 4, OUTD / 64), 128, 0, stream>>>(
      x1, Wsk, out, Mt_nodes, HID, HID, OUTD, OUTD);
  final_add<<<N, OUTD, 0, stream>>>(out, xp2, cid, bsk, N);
}